// GCNRegressor_29721173689112
// MI455X (gfx1250) — compile-verified
//
#include <hip/hip_runtime.h>
#include <hip/hip_bf16.h>

#define H 64
#define BN_EPS 1e-5f

typedef float v2f __attribute__((ext_vector_type(2)));
typedef float v8f __attribute__((ext_vector_type(8)));

// ================= CSR build (once, reused by all 4 layers) =================
__global__ void k_zero_cnt(int* cnt, int N) {
    int i = blockIdx.x * blockDim.x + threadIdx.x;
    if (i < N) cnt[i] = 0;
}
__global__ void k_cnt(const int* __restrict__ ei, int* cnt, int E) {
    int e = blockIdx.x * blockDim.x + threadIdx.x;
    if (e < E) atomicAdd(&cnt[ei[E + e]], 1);
}
// dinv = 1/sqrt(deg), deg = in-count + 1 (self-loop)
__global__ void k_dinv(const int* __restrict__ cnt, float* dinv, int N) {
    int i = blockIdx.x * blockDim.x + threadIdx.x;
    if (i < N) dinv[i] = rsqrtf((float)cnt[i] + 1.0f);
}
// Single-block exclusive scan (Hillis-Steele per 1024-tile with running carry).
__global__ void __launch_bounds__(1024)
k_scan_csr(const int* __restrict__ cnt, int* rowptr, int* cursor, int N) {
    __shared__ int sm[1024];
    __shared__ int carry;
    const int tid = threadIdx.x;
    if (tid == 0) carry = 0;
    __syncthreads();
    for (int base = 0; base < N; base += 1024) {
        int i = base + tid;
        int v = (i < N) ? cnt[i] : 0;
        sm[tid] = v;
        __syncthreads();
#pragma unroll
        for (int off = 1; off < 1024; off <<= 1) {
            int t = (tid >= off) ? sm[tid - off] : 0;
            __syncthreads();
            sm[tid] += t;
            __syncthreads();
        }
        int base_carry = carry;                  // prefix before this tile
        if (i < N) {
            int r = base_carry + sm[tid] - v;    // exclusive
            rowptr[i] = r;
            cursor[i] = r;
        }
        __syncthreads();
        if (tid == 0) carry = base_carry + sm[1023];
        __syncthreads();
    }
    if (tid == 0) rowptr[N] = carry;
}
__global__ void k_csr_fill(const int* __restrict__ ei, int* cursor, int* col, int E) {
    int e = blockIdx.x * blockDim.x + threadIdx.x;
    if (e >= E) return;
    int d = ei[E + e];
    int pos = atomicAdd(&cursor[d], 1);
    col[pos] = ei[e];  // source node of an in-edge of d
}

// ========== WMMA f32 GEMM: Out[N x 64] = A[N x K] @ B[K x 64] ==========
// 128 threads = 4 waves; block owns a 16-row strip, wave w owns cols [16w,16w+16).
// In-place (Out==A) safe: __syncthreads() separates all A reads from stores.
__global__ void __launch_bounds__(128)
k_gemm_wmma(const float* __restrict__ A, const float* __restrict__ Bm,
            const float* __restrict__ bias, float* __restrict__ Out,
            int N, int K, int fuse_relu) {
    const int lane = threadIdx.x & 31;
    const int wave = threadIdx.x >> 5;
    const int m0   = blockIdx.x * 16;
    const int n0   = wave * 16;
    const int half = lane >> 4;
    const int l16  = lane & 15;
    const int koff = half * 2;  // ISA 16x4 f32 A/B fragment layout

    int arow = m0 + l16;
    if (arow >= N) arow = N - 1;
    const float* __restrict__ ap = A + (size_t)arow * K;

    v8f c = {};
    for (int kk = 0; kk < K; kk += 4) {
        v2f a, b;
        a.x = ap[kk + koff];
        a.y = ap[kk + koff + 1];
        b.x = Bm[(size_t)(kk + koff) * H + n0 + l16];
        b.y = Bm[(size_t)(kk + koff + 1) * H + n0 + l16];
        c = __builtin_amdgcn_wmma_f32_16x16x4_f32(false, a, false, b,
                                                  (short)0, c, false, false);
    }
    __syncthreads();

    const int ocol = n0 + l16;
    const float bv = bias ? bias[ocol] : 0.0f;
    const int orow0 = m0 + half * 8;
#pragma unroll
    for (int r = 0; r < 8; ++r) {
        int orow = orow0 + r;
        if (orow < N) {
            float v = c[r] + bv;
            if (fuse_relu) v = fmaxf(v, 0.0f);
            Out[(size_t)orow * H + ocol] = v;
        }
    }
}

// ========== gather aggregation: agg[d] = dn*(sum dinv[s]*hw[s] + dn*hw[d]) + b ==========
// 16 threads per node, float4 per thread (coalesced 256B row reads). No atomics.
__global__ void k_aggregate(const float* __restrict__ hw, const int* __restrict__ rowptr,
                            const int* __restrict__ col, const float* __restrict__ dinv,
                            const float* __restrict__ bias, float* __restrict__ outp,
                            int N) {
    int t = blockIdx.x * blockDim.x + threadIdx.x;
    int node = t >> 4;
    int c4 = (t & 15) * 4;
    if (node >= N) return;
    const int beg = rowptr[node];
    const int end = rowptr[node + 1];
    const float dn = dinv[node];

    float4 acc = make_float4(0.f, 0.f, 0.f, 0.f);
    for (int j = beg; j < end; ++j) {
        int s = col[j];
        if (j + 4 < end) {  // warm L2/L1 for an upcoming source row
            __builtin_prefetch(hw + (size_t)col[j + 4] * H + c4, 0, 1);
        }
        float w = dinv[s];
        float4 hv = *(const float4*)(hw + (size_t)s * H + c4);
        acc.x += hv.x * w; acc.y += hv.y * w;
        acc.z += hv.z * w; acc.w += hv.w * w;
    }
    float4 hs = *(const float4*)(hw + (size_t)node * H + c4);
    float4 r;
    r.x = (acc.x + hs.x * dn) * dn + bias[c4 + 0];
    r.y = (acc.y + hs.y * dn) * dn + bias[c4 + 1];
    r.z = (acc.z + hs.z * dn) * dn + bias[c4 + 2];
    r.w = (acc.w + hs.w * dn) * dn + bias[c4 + 3];
    *(float4*)(outp + (size_t)node * H + c4) = r;
}

// ================= batch norm =================
__global__ void k_zero_stats(float* stats) { stats[threadIdx.x] = 0.0f; }  // 128 thr
__global__ void k_bn_reduce(const float* __restrict__ h, float* stats, int N) {
    int ch = threadIdx.x & 63;
    int row0 = blockIdx.x * (blockDim.x >> 6) + (threadIdx.x >> 6);
    int stride = gridDim.x * (blockDim.x >> 6);
    float s = 0.0f, s2 = 0.0f;
    for (int n = row0; n < N; n += stride) {
        float v = h[(size_t)n * H + ch];
        s += v;
        s2 += v * v;
    }
    atomicAdd(&stats[ch], s);
    atomicAdd(&stats[H + ch], s2);
}
__global__ void k_bn_apply(const float* __restrict__ hin, const float* __restrict__ stats,
                           const float* __restrict__ gamma, const float* __restrict__ beta,
                           float* __restrict__ hout, int N) {
    int q = blockIdx.x * blockDim.x + threadIdx.x;  // one float4 per thread
    if (q >= N * (H / 4)) return;
    int c4 = (q & 15) * 4;
    float inv_n = 1.0f / (float)N;
    float4 v = *(const float4*)(hin + (size_t)q * 4);
    float4 r;
#pragma unroll
    for (int k = 0; k < 4; ++k) {
        int ch = c4 + k;
        float mean = stats[ch] * inv_n;
        float var = stats[H + ch] * inv_n - mean * mean;  // biased var
        float sc = gamma[ch] * rsqrtf(var + BN_EPS);
        float val = ((&v.x)[k] - mean) * sc + beta[ch];
        (&r.x)[k] = fmaxf(val, 0.0f);  // fused ReLU
    }
    *(float4*)(hout + (size_t)q * 4) = r;
}

// ========== head projection [64 -> 1], one wave32 per node ==========
__global__ void k_head_dot(const float* __restrict__ h, const float* __restrict__ w2,
                           const float* __restrict__ b2, float* __restrict__ out, int N) {
    int gid = blockIdx.x * blockDim.x + threadIdx.x;
    int node = gid >> 5;
    int lane = threadIdx.x & 31;
    if (node >= N) return;
    float v = h[(size_t)node * H + lane] * w2[lane] +
              h[(size_t)node * H + 32 + lane] * w2[32 + lane];
#pragma unroll
    for (int o = 16; o > 0; o >>= 1) v += __shfl_down(v, o, 32);
    if (lane == 0) out[node] = v + b2[0];
}

extern "C" void kernel_launch(void* const* d_in, const int* in_sizes, int n_in,
                              void* d_out, int out_size, void* d_ws, size_t ws_size,
                              hipStream_t stream) {
    const float* x      = (const float*)d_in[0];   // [N,128]
    const int*   ei     = (const int*)d_in[1];     // [2,E]
    const float* W0     = (const float*)d_in[2];   // [128,64]
    const float* b0     = (const float*)d_in[3];   // [64]
    const float* Ws     = (const float*)d_in[4];   // [3,64,64]
    const float* bs     = (const float*)d_in[5];   // [3,64]
    const float* gammas = (const float*)d_in[6];   // [4,64]
    const float* betas  = (const float*)d_in[7];   // [4,64]
    const float* hW1    = (const float*)d_in[8];   // [64,64]
    const float* hb1    = (const float*)d_in[9];   // [64]
    const float* hW2    = (const float*)d_in[10];  // [64,1]
    const float* hb2    = (const float*)d_in[11];  // [1]
    float* out = (float*)d_out;

    const int N = in_sizes[0] / 128;  // 100000
    const int E = in_sizes[1] / 2;    // 1600000
    const int NH = N * H;

    // workspace carve-up (floats/ints, 512B-aligned sections)
    char* base = (char*)d_ws;
    size_t off = 0;
    auto carve = [&](size_t bytes) {
        char* p = base + off;
        off = (off + bytes + 511) & ~(size_t)511;
        return p;
    };
    float* dinv   = (float*)carve((size_t)N * 4);
    float* bufA   = (float*)carve((size_t)NH * 4);
    float* bufB   = (float*)carve((size_t)NH * 4);
    float* stats  = (float*)carve(128 * 4);
    int*   cnt    = (int*)carve((size_t)N * 4);
    int*   rowptr = (int*)carve((size_t)(N + 1) * 4);
    int*   cursor = (int*)carve((size_t)N * 4);
    int*   col    = (int*)carve((size_t)E * 4);

    // ---- CSR-by-destination build + symmetric norm (once) ----
    k_zero_cnt<<<(N + 255) / 256, 256, 0, stream>>>(cnt, N);
    k_cnt<<<(E + 255) / 256, 256, 0, stream>>>(ei, cnt, E);
    k_dinv<<<(N + 255) / 256, 256, 0, stream>>>(cnt, dinv, N);
    k_scan_csr<<<1, 1024, 0, stream>>>(cnt, rowptr, cursor, N);
    k_csr_fill<<<(E + 255) / 256, 256, 0, stream>>>(ei, cursor, col, E);

    dim3 ggrid((N + 15) / 16), gblock(128);
    int agg_blocks = (N * 16 + 255) / 256;

    for (int layer = 0; layer < 4; ++layer) {
        const float* Hin = (layer == 0) ? x : bufA;
        const int    K   = (layer == 0) ? 128 : H;
        const float* W   = (layer == 0) ? W0 : (Ws + (size_t)(layer - 1) * H * H);
        const float* b   = (layer == 0) ? b0 : (bs + (size_t)(layer - 1) * H);

        // linear transform via WMMA (in-place safe for layers > 0)
        k_gemm_wmma<<<ggrid, gblock, 0, stream>>>(Hin, W, nullptr, bufA, N, K, 0);
        // atomic-free gather aggregation (norm + self-loop + bias fused)
        k_aggregate<<<agg_blocks, 256, 0, stream>>>(bufA, rowptr, col, dinv, b, bufB, N);
        // batch norm (training-mode batch stats) + ReLU -> bufA
        k_zero_stats<<<1, 128, 0, stream>>>(stats);
        k_bn_reduce<<<512, 256, 0, stream>>>(bufB, stats, N);
        k_bn_apply<<<(NH / 4 + 255) / 256, 256, 0, stream>>>(
            bufB, stats, gammas + layer * H, betas + layer * H, bufA, N);
    }

    // MLP head: relu(h @ hW1 + hb1) in-place, then [64->1] projection
    k_gemm_wmma<<<ggrid, gblock, 0, stream>>>(bufA, hW1, hb1, bufA, N, H, 1);
    k_head_dot<<<(N * 32 + 255) / 256, 256, 0, stream>>>(bufA, hW2, hb2, out, N);
}